// GaussianPolicy_14620068675625
// MI455X (gfx1250) — compile-verified
//
#include <hip/hip_runtime.h>

typedef __attribute__((ext_vector_type(2))) float v2f;
typedef __attribute__((ext_vector_type(8))) float v8f;

#define DECAY  0.2f
#define THRESH 0.2f
#define LOGS_MIN (-20.0f)
#define LOGS_MAX (2.0f)

#define BDIM   4096
#define HDIM   2048
#define INDIM  512
#define ADIM   32

#define KC       64              // K-chunk staged in LDS (floats)
#define LDS_PAD  68              // row stride in dwords: 68%64==4 -> conflict-free,
                                 // 68*4B=272B=17*16B -> rows stay 16B-aligned

__device__ __forceinline__ v8f wmma4(v2f a, v2f b, v8f c) {
    // D = A(16x4 f32) * B(4x16 f32) + C(16x16 f32), wave32
    return __builtin_amdgcn_wmma_f32_16x16x4_f32(
        /*neg_a=*/false, a, /*neg_b=*/false, b,
        /*c_mod=*/(short)0, c, /*reuse_a=*/false, /*reuse_b=*/false);
}

__device__ __forceinline__ v2f ld2(const float* p) {
    v2f r; r.x = p[0]; r.y = p[1]; return r;
}

__device__ __forceinline__ v8f vzero8() {
    v8f z = {0.f, 0.f, 0.f, 0.f, 0.f, 0.f, 0.f, 0.f};
    return z;
}

// Pointer types the async builtin expects: int4-vector pointers in AS1 / AS3.
typedef int v4i_ __attribute__((vector_size(16)));
typedef __attribute__((address_space(1))) v4i_* g128p;   // global
typedef __attribute__((address_space(3))) v4i_* l128p;   // LDS

// 16-byte global -> LDS copy. Prefer CDNA5 async path (ASYNCcnt-tracked DMA
// into LDS); fall back to a regular load+store if the builtin is unavailable.
// Casts go through integers: generic global ptr == 64-bit global address;
// low 32 bits of a generic LDS ptr == LDS byte offset (ISA 10.2).
__device__ __forceinline__ void copy16_g2l(const float* gp, float* lp) {
#if __has_builtin(__builtin_amdgcn_global_load_async_to_lds_b128)
    __builtin_amdgcn_global_load_async_to_lds_b128(
        (g128p)(unsigned long long)gp,
        (l128p)(unsigned int)(unsigned long long)lp,
        /*offset=*/0, /*cpol=*/0);
#else
    *(float4*)lp = *(const float4*)gp;
#endif
}

__device__ __forceinline__ void async_copies_join() {
#if __has_builtin(__builtin_amdgcn_global_load_async_to_lds_b128)
#if __has_builtin(__builtin_amdgcn_s_wait_asynccnt)
    __builtin_amdgcn_s_wait_asynccnt(0);
#else
    asm volatile("s_wait_asynccnt 0x0" ::: "memory");
#endif
#endif
}

// ---------------------------------------------------------------------------
// Kernel 1: fused  fc = state_t @ W_lif^T + b_lif  (5 unique timesteps)
//           + 15-step LIF scan kept in registers -> x = mean(spikes)
// Wave tile 32x32, block = 8 waves -> 64 (M) x 128 (N) block tile.
// ---------------------------------------------------------------------------
__global__ __launch_bounds__(256) void lif_fc_kernel(
    const float* __restrict__ state,   // [4096, 5, 512]
    const float* __restrict__ Wl,      // [2048, 512]
    const float* __restrict__ bl,      // [2048]
    float* __restrict__ xout)          // [4096, 2048]
{
    const int lane = threadIdx.x & 31;
    const int wave = threadIdx.x >> 5;
    const int idx  = lane & 15;
    const int half = lane >> 4;
    const int m0 = blockIdx.y * 64  + (wave >> 2) * 32;
    const int n0 = blockIdx.x * 128 + (wave & 3)  * 32;

    const float* bp[2] = { Wl + (size_t)(n0 + idx)      * INDIM,
                           Wl + (size_t)(n0 + 16 + idx) * INDIM };
    const float biasj[2] = { bl[n0 + idx], bl[n0 + 16 + idx] };

    v8f mem[2][2], spk[2][2], acc[2][2];
#pragma unroll
    for (int i = 0; i < 2; ++i)
#pragma unroll
        for (int j = 0; j < 2; ++j) {
            mem[i][j] = vzero8(); spk[i][j] = vzero8(); acc[i][j] = vzero8();
        }

    for (int t = 0; t < 5; ++t) {
        const float* ap[2] = {
            state + ((size_t)(m0 + idx)      * 5 + t) * INDIM,
            state + ((size_t)(m0 + 16 + idx) * 5 + t) * INDIM };

        v8f f[2][2];
#pragma unroll
        for (int i = 0; i < 2; ++i)
#pragma unroll
            for (int j = 0; j < 2; ++j) f[i][j] = vzero8();

#pragma unroll 4
        for (int k = 0; k < INDIM; k += 4) {
            const int kk = k + 2 * half;
            v2f a[2] = { ld2(ap[0] + kk), ld2(ap[1] + kk) };
            v2f b[2] = { ld2(bp[0] + kk), ld2(bp[1] + kk) };
#pragma unroll
            for (int i = 0; i < 2; ++i)
#pragma unroll
                for (int j = 0; j < 2; ++j)
                    f[i][j] = wmma4(a[i], b[j], f[i][j]);
        }

        // LIF update: each fc_t is applied 3x (jnp.repeat(state,3,axis=1))
#pragma unroll
        for (int i = 0; i < 2; ++i)
#pragma unroll
            for (int j = 0; j < 2; ++j)
#pragma unroll
                for (int r = 0; r < 8; ++r) {
                    const float fc = f[i][j][r] + biasj[j];
                    float m = mem[i][j][r];
                    float s = spk[i][j][r];
                    float a = acc[i][j][r];
#pragma unroll
                    for (int rep = 0; rep < 3; ++rep) {
                        m = m * DECAY * (1.0f - s) + fc;
                        s = (m - THRESH) > 0.0f ? 1.0f : 0.0f;
                        a += s;
                    }
                    mem[i][j][r] = m; spk[i][j][r] = s; acc[i][j][r] = a;
                }
    }

    // x = mean over 15 steps
#pragma unroll
    for (int i = 0; i < 2; ++i)
#pragma unroll
        for (int j = 0; j < 2; ++j)
#pragma unroll
            for (int r = 0; r < 8; ++r) {
                const int m = m0 + i * 16 + r + 8 * half;
                const int n = n0 + j * 16 + idx;
                xout[(size_t)m * HDIM + n] = acc[i][j][r] * (1.0f / 15.0f);
            }
}

// ---------------------------------------------------------------------------
// Kernel 2: out[4096,2048] = relu(A[4096,K] @ W[2048,K]^T + bias)
// LDS-staged: async global->LDS copies of 64x64 A tile + 128x64 B tile per
// K-chunk; WMMA fragments fed by ds_load_b64 (bank-conflict-free padding).
// ---------------------------------------------------------------------------
__global__ __launch_bounds__(256) void gemm_relu_kernel(
    const float* __restrict__ A,
    const float* __restrict__ W,
    const float* __restrict__ bias,
    float* __restrict__ out,
    int K)
{
    __shared__ float lA[64  * LDS_PAD];
    __shared__ float lB[128 * LDS_PAD];

    const int tid  = threadIdx.x;
    const int lane = tid & 31;
    const int wave = tid >> 5;
    const int idx  = lane & 15;
    const int half = lane >> 4;
    const int wm   = (wave >> 2) * 32;           // wave M offset in block tile
    const int wn   = (wave & 3)  * 32;           // wave N offset in block tile
    const int mblk = blockIdx.y * 64;
    const int nblk = blockIdx.x * 128;

    const float biasj[2] = { bias[nblk + wn + idx], bias[nblk + wn + 16 + idx] };

    v8f c[2][2];
#pragma unroll
    for (int i = 0; i < 2; ++i)
#pragma unroll
        for (int j = 0; j < 2; ++j) c[i][j] = vzero8();

    for (int k0 = 0; k0 < K; k0 += KC) {
        // --- stage A tile: 64 rows x 16 segs(16B) = 1024 segs, 4/thread ---
#pragma unroll
        for (int s = 0; s < 4; ++s) {
            const int seg = tid + s * 256;
            const int row = seg >> 4;
            const int c4  = (seg & 15) * 4;
            copy16_g2l(A + (size_t)(mblk + row) * K + k0 + c4,
                       lA + row * LDS_PAD + c4);
        }
        // --- stage B tile: 128 rows x 16 segs = 2048 segs, 8/thread ---
#pragma unroll
        for (int s = 0; s < 8; ++s) {
            const int seg = tid + s * 256;
            const int row = seg >> 4;
            const int c4  = (seg & 15) * 4;
            copy16_g2l(W + (size_t)(nblk + row) * K + k0 + c4,
                       lB + row * LDS_PAD + c4);
        }
        async_copies_join();
        __syncthreads();

        const float* la[2] = { lA + (wm + idx)      * LDS_PAD,
                               lA + (wm + 16 + idx) * LDS_PAD };
        const float* lb[2] = { lB + (wn + idx)      * LDS_PAD,
                               lB + (wn + 16 + idx) * LDS_PAD };
#pragma unroll 4
        for (int k = 0; k < KC; k += 4) {
            const int kk = k + 2 * half;
            v2f a[2] = { ld2(la[0] + kk), ld2(la[1] + kk) };
            v2f b[2] = { ld2(lb[0] + kk), ld2(lb[1] + kk) };
#pragma unroll
            for (int i = 0; i < 2; ++i)
#pragma unroll
                for (int j = 0; j < 2; ++j)
                    c[i][j] = wmma4(a[i], b[j], c[i][j]);
        }
        __syncthreads();
    }

#pragma unroll
    for (int i = 0; i < 2; ++i)
#pragma unroll
        for (int j = 0; j < 2; ++j)
#pragma unroll
            for (int r = 0; r < 8; ++r) {
                const int m = mblk + wm + i * 16 + r + 8 * half;
                const int n = nblk + wn + j * 16 + idx;
                out[(size_t)m * HDIM + n] = fmaxf(c[i][j][r] + biasj[j], 0.0f);
            }
}

// ---------------------------------------------------------------------------
// Kernel 3: heads, N = 32.  out[4096,32] = A[4096,2048] @ W[32,2048]^T + bias
// Block = 8 waves stacked in M -> 256 (M) x 32 (N) block tile.
// ---------------------------------------------------------------------------
__global__ __launch_bounds__(256) void head_kernel(
    const float* __restrict__ A,
    const float* __restrict__ W,
    const float* __restrict__ bias,
    float* __restrict__ out,
    int do_clip)
{
    const int lane = threadIdx.x & 31;
    const int wave = threadIdx.x >> 5;
    const int idx  = lane & 15;
    const int half = lane >> 4;
    const int m0 = blockIdx.x * 256 + wave * 32;

    const float* ap[2] = { A + (size_t)(m0 + idx)      * HDIM,
                           A + (size_t)(m0 + 16 + idx) * HDIM };
    const float* bp[2] = { W + (size_t)(idx)      * HDIM,
                           W + (size_t)(16 + idx) * HDIM };
    const float biasj[2] = { bias[idx], bias[16 + idx] };

    v8f c[2][2];
#pragma unroll
    for (int i = 0; i < 2; ++i)
#pragma unroll
        for (int j = 0; j < 2; ++j) c[i][j] = vzero8();

#pragma unroll 4
    for (int k = 0; k < HDIM; k += 4) {
        const int kk = k + 2 * half;
        v2f a[2] = { ld2(ap[0] + kk), ld2(ap[1] + kk) };
        v2f b[2] = { ld2(bp[0] + kk), ld2(bp[1] + kk) };
#pragma unroll
        for (int i = 0; i < 2; ++i)
#pragma unroll
            for (int j = 0; j < 2; ++j)
                c[i][j] = wmma4(a[i], b[j], c[i][j]);
    }

#pragma unroll
    for (int i = 0; i < 2; ++i)
#pragma unroll
        for (int j = 0; j < 2; ++j)
#pragma unroll
            for (int r = 0; r < 8; ++r) {
                const int m = m0 + i * 16 + r + 8 * half;
                const int n = j * 16 + idx;
                float v = c[i][j][r] + biasj[j];
                if (do_clip) v = fminf(fmaxf(v, LOGS_MIN), LOGS_MAX);
                out[(size_t)m * ADIM + n] = v;
            }
}

// ---------------------------------------------------------------------------
extern "C" void kernel_launch(void* const* d_in, const int* in_sizes, int n_in,
                              void* d_out, int out_size, void* d_ws, size_t ws_size,
                              hipStream_t stream) {
    (void)in_sizes; (void)n_in; (void)out_size; (void)ws_size;

    const float* state = (const float*)d_in[0];
    const float* Wl  = (const float*)d_in[1];
    const float* bl  = (const float*)d_in[2];
    const float* W11 = (const float*)d_in[3];
    const float* b11 = (const float*)d_in[4];
    const float* W12 = (const float*)d_in[5];
    const float* b12 = (const float*)d_in[6];
    const float* W21 = (const float*)d_in[7];
    const float* b21 = (const float*)d_in[8];
    const float* W22 = (const float*)d_in[9];
    const float* b22 = (const float*)d_in[10];
    const float* Wm  = (const float*)d_in[11];
    const float* bm  = (const float*)d_in[12];
    const float* Wls = (const float*)d_in[13];
    const float* bls = (const float*)d_in[14];

    float* out = (float*)d_out;
    float* ws  = (float*)d_ws;

    const size_t BUF = (size_t)BDIM * HDIM;   // 8M floats = 32 MB
    float* x  = ws;                           // LIF output
    float* x1 = ws + BUF;                     // branch-1 hidden 1
    float* x2 = ws + 2 * BUF;                 // branch-2 hidden 1
    float* y1 = ws + 3 * BUF;                 // branch-1 hidden 2
    float* y2 = ws;                           // reuse x's buffer (x dead by then)

    dim3 blk(256);
    dim3 grd(HDIM / 128, BDIM / 64);          // (16, 64)

    lif_fc_kernel   <<<grd, blk, 0, stream>>>(state, Wl, bl, x);
    gemm_relu_kernel<<<grd, blk, 0, stream>>>(x,  W11, b11, x1, HDIM);
    gemm_relu_kernel<<<grd, blk, 0, stream>>>(x,  W21, b21, x2, HDIM);
    gemm_relu_kernel<<<grd, blk, 0, stream>>>(x1, W12, b12, y1, HDIM);
    gemm_relu_kernel<<<grd, blk, 0, stream>>>(x2, W22, b22, y2, HDIM);

    dim3 hgrd(BDIM / 256);                    // 16 blocks
    head_kernel<<<hgrd, blk, 0, stream>>>(y1, Wm,  bm,  out,                       0);
    head_kernel<<<hgrd, blk, 0, stream>>>(y2, Wls, bls, out + (size_t)BDIM * ADIM, 1);
}